// LinkLevelGNN_18459769438574
// MI455X (gfx1250) — compile-verified
//
#include <hip/hip_runtime.h>

typedef __attribute__((ext_vector_type(2))) float v2f;
typedef __attribute__((ext_vector_type(8))) float v8f;

// ---------------------------------------------------------------------------
// Degree: deg[dst] += 1 per edge (self-loop handled as +1 in dinv kernel)
// ---------------------------------------------------------------------------
__global__ void degree_kernel(const long long* __restrict__ dst,
                              float* __restrict__ deg, int E) {
  int e = blockIdx.x * blockDim.x + threadIdx.x;
  if (e < E) atomicAdd(&deg[(int)dst[e]], 1.0f);
}

__global__ void dinv_kernel(const float* __restrict__ deg,
                            float* __restrict__ dinv, int n) {
  int i = blockIdx.x * blockDim.x + threadIdx.x;
  if (i < n) dinv[i] = rsqrtf(deg[i] + 1.0f);   // self-loop makes deg >= 1
}

// ---------------------------------------------------------------------------
// C[M x 128] = A[M x K] * B[K x 128], full fp32 via V_WMMA_F32_16X16X4_F32.
// Block = 256 threads = 8 waves. Block handles one 16-row M tile (staged in
// LDS); wave w handles output columns [16w, 16w+16). M must be multiple of 16.
//
// Fragment layouts (CDNA5 ISA 7.12.2, wave32):
//   A 16x4 f32 : lane L -> row L%16, VGPR e -> K = k0 + 2*(L/16) + e
//   B 4x16 f32 : lane L -> col L%16, VGPR e -> K = k0 + 2*(L/16) + e
//   C 16x16 f32: lane L -> col L%16, VGPR r -> row r + 8*(L/16)
// ---------------------------------------------------------------------------
template <int K>
__global__ void gemm_wmma_f32_n128(const float* __restrict__ A,
                                   const float* __restrict__ B,
                                   float* __restrict__ C) {
  __shared__ float As[16 * K];
  const int tid = threadIdx.x;
  const long long tileM = blockIdx.x;

  // Cooperative stage of the 16 x K A tile (float4 vectorized).
  const float* Ablk = A + tileM * 16 * K;
  constexpr int n4 = (16 * K) / 4;
  for (int i = tid; i < n4; i += 256)
    ((float4*)As)[i] = ((const float4*)Ablk)[i];
  __syncthreads();

  const int wave = tid >> 5;   // 0..7 -> N tile
  const int lane = tid & 31;
  const int r16  = lane & 15;  // A row / B,C column within tile
  const int half = lane >> 4;  // 0 or 1

  v8f acc = {};
  const float* Bp = B + wave * 16 + r16;  // column base in B

#pragma unroll 8
  for (int k0 = 0; k0 < K; k0 += 4) {
    const int ka = k0 + half * 2;
    v2f a = *(const v2f*)(As + r16 * K + ka);       // aligned 8B LDS read
    v2f b;
    b.x = Bp[(long long)ka * 128];
    b.y = Bp[(long long)(ka + 1) * 128];
    acc = __builtin_amdgcn_wmma_f32_16x16x4_f32(
        false, a, false, b, (short)0, acc, false, false);
  }

  float* Cp = C + (tileM * 16) * 128 + wave * 16 + r16;
#pragma unroll
  for (int r = 0; r < 8; ++r)
    Cp[(long long)(r + half * 8) * 128] = acc[r];
}

// ---------------------------------------------------------------------------
// out[dst] += dinv[src]*dinv[dst] * h[src]   (32 threads per edge, 4 ch each)
// ---------------------------------------------------------------------------
__global__ void aggregate_kernel(const float* __restrict__ h,
                                 const long long* __restrict__ src,
                                 const long long* __restrict__ dst,
                                 const float* __restrict__ dinv,
                                 float* __restrict__ out, int E) {
  long long t = (long long)blockIdx.x * blockDim.x + threadIdx.x;
  int e = (int)(t >> 5);
  int g = (int)(t & 31);
  if (e >= E) return;
  int s = (int)src[e];
  int d = (int)dst[e];
  float nrm = dinv[s] * dinv[d];
  float4 hv = *(const float4*)(h + (long long)s * 128 + g * 4);
  float* o = out + (long long)d * 128 + g * 4;
  atomicAdd(o + 0, nrm * hv.x);
  atomicAdd(o + 1, nrm * hv.y);
  atomicAdd(o + 2, nrm * hv.z);
  atomicAdd(o + 3, nrm * hv.w);
}

// a[i] = relu(a[i] + dinv[node]^2 * h[i] + bias[c])   (self-loop + bias + relu)
__global__ void epilogue_relu(float* __restrict__ a, const float* __restrict__ h,
                              const float* __restrict__ dinv,
                              const float* __restrict__ bias, int total) {
  int i = blockIdx.x * blockDim.x + threadIdx.x;
  if (i >= total) return;
  int node = i >> 7;
  int c = i & 127;
  float di = dinv[node];
  float v = a[i] + di * di * h[i] + bias[c];
  a[i] = fmaxf(v, 0.0f);
}

// o[i] += dinv[node]^2 * h[i] + bias[c]
__global__ void epilogue_out(float* __restrict__ o, const float* __restrict__ h,
                             const float* __restrict__ dinv,
                             const float* __restrict__ bias, int total) {
  int i = blockIdx.x * blockDim.x + threadIdx.x;
  if (i >= total) return;
  int node = i >> 7;
  int c = i & 127;
  float di = dinv[node];
  o[i] = o[i] + di * di * h[i] + bias[c];
}

// ---------------------------------------------------------------------------
extern "C" void kernel_launch(void* const* d_in, const int* in_sizes, int n_in,
                              void* d_out, int out_size, void* d_ws, size_t ws_size,
                              hipStream_t stream) {
  const float*     x  = (const float*)d_in[0];
  const long long* ei = (const long long*)d_in[1];  // int64 edge_index [2, E]
  const float*     W1 = (const float*)d_in[2];
  const float*     b1 = (const float*)d_in[3];
  const float*     W2 = (const float*)d_in[4];
  const float*     b2 = (const float*)d_in[5];
  float* out = (float*)d_out;
  (void)n_in; (void)out_size; (void)ws_size;

  const int HID = in_sizes[3];           // 128
  const int INC = in_sizes[2] / HID;     // 384
  const int N   = in_sizes[0] / INC;     // 50000 (multiple of 16)
  const int E   = in_sizes[1] / 2;       // 800000
  (void)HID;

  // Workspace layout (floats): deg[N] | dinv[N] | bufH[N*128] | bufA[N*128]
  float* deg  = (float*)d_ws;
  float* dinv = deg + N;
  float* bufH = dinv + N;                       // h1, later h2
  float* bufA = bufH + (size_t)N * 128;         // agg1, later a1 (relu'd)

  const long long* src = ei;
  const long long* dst = ei + E;

  const int total = N * 128;
  hipMemsetAsync(deg,  0, (size_t)N * sizeof(float), stream);
  hipMemsetAsync(bufA, 0, (size_t)total * sizeof(float), stream);
  hipMemsetAsync(out,  0, (size_t)total * sizeof(float), stream);

  degree_kernel<<<(E + 255) / 256, 256, 0, stream>>>(dst, deg, E);
  dinv_kernel<<<(N + 255) / 256, 256, 0, stream>>>(deg, dinv, N);

  // Layer 1: h1 = x @ W1 ; agg ; + self-loop + b1 ; relu
  gemm_wmma_f32_n128<384><<<N / 16, 256, 0, stream>>>(x, W1, bufH);
  {
    long long aggT = (long long)E * 32;
    aggregate_kernel<<<(int)((aggT + 255) / 256), 256, 0, stream>>>(
        bufH, src, dst, dinv, bufA, E);
  }
  epilogue_relu<<<(total + 255) / 256, 256, 0, stream>>>(bufA, bufH, dinv, b1, total);

  // Layer 2: h2 = a1 @ W2 ; agg into d_out ; + self-loop + b2
  gemm_wmma_f32_n128<128><<<N / 16, 256, 0, stream>>>(bufA, W2, bufH);
  {
    long long aggT = (long long)E * 32;
    aggregate_kernel<<<(int)((aggT + 255) / 256), 256, 0, stream>>>(
        bufH, src, dst, dinv, out, E);
  }
  epilogue_out<<<(total + 255) / 256, 256, 0, stream>>>(out, bufH, dinv, b2, total);
}